// VideoModel_56032143343991
// MI455X (gfx1250) — compile-verified
//
#include <hip/hip_runtime.h>
#include <hip/hip_bf16.h>

typedef __attribute__((ext_vector_type(2))) float v2f;
typedef __attribute__((ext_vector_type(8))) float v8f;

#define BATCH 16
#define CIN   32

// SWAPX16 lane exchange (wave32): ds_swizzle group-of-32, xor=0x10, and=0x1f
__device__ __forceinline__ float xor16(float v) {
    return __int_as_float(__builtin_amdgcn_ds_swizzle(__float_as_int(v), 0x401F));
}

// ---------------------------------------------------------------------------
// 3x3 SAME conv as im2col GEMM on V_WMMA_F32_16X16X4_F32.
//   out[Cout,16q] tile = W[Cout,288] x patches[288,16q]
// K enumerated tap-major: k' = r*32 + i  (r = 3x3 tap, i = input channel), so
// tap geometry/border logic is hoisted out of the 8-chunk inner loop and all
// gathers are unconditional loads from clamped addresses scaled by a 0/1
// validity factor (no exec-mask divergence, no integer division).
// One wave32 per (batch, 16-query tile). Cout <= 16; rows >= Cout are zero.
// ---------------------------------------------------------------------------
__global__ void __launch_bounds__(256)
conv3x3_wmma(const float* __restrict__ in, const float* __restrict__ w,
             const float* __restrict__ bias, float* __restrict__ out,
             int Cout, int H, int W, int outBatchStride, int nQT)
{
    const int lane = threadIdx.x & 31;
    const int wave = threadIdx.x >> 5;
    const int qt   = blockIdx.x * 8 + wave;
    const int b    = blockIdx.y;
    if (qt >= nQT) return;                 // wave-uniform exit

    const int N   = H * W;
    const int q0  = qt * 16;
    const int col = lane & 15;
    const int hi  = lane >> 4;

    int q = q0 + col; if (q > N - 1) q = N - 1;   // clamp tail columns
    const int y = q / W, x = q - y * W;
    const float* inb = in + (size_t)b * CIN * N;

    // A operand rows are output channels; lane feeds row M=col.
    // Clamp the address for rows >= Cout and zero them via a 0/1 factor.
    const int   colc = col < Cout ? col : 0;
    const float aw   = col < Cout ? 1.0f : 0.0f;
    const float* wrow = w + (size_t)colc * (CIN * 9);

    v8f acc = {};
#pragma unroll
    for (int r = 0; r < 9; ++r) {          // 3x3 tap (unrolled: constants)
        const int ky = r / 3, kx = r - (r / 3) * 3;
        const int yy = y + ky - 1, xx = x + kx - 1;
        const bool ok  = (yy >= 0) && (yy < H) && (xx >= 0) && (xx < W);
        const float okf = ok ? 1.0f : 0.0f;
        const int yyc = yy < 0 ? 0 : (yy > H - 1 ? H - 1 : yy);
        const int xxc = xx < 0 ? 0 : (xx > W - 1 ? W - 1 : xx);
        const int pix = yyc * W + xxc;     // always-valid pixel address
        const float* ip = inb + pix;
#pragma unroll
        for (int cc = 0; cc < 8; ++cc) {   // channel chunks of 4
            const int i = cc * 4 + 2 * hi;
            v2f a;
            a.x = wrow[(i + 0) * 9 + r] * aw;
            a.y = wrow[(i + 1) * 9 + r] * aw;
            v2f bb;
            bb.x = ip[(size_t)(i + 0) * N] * okf;
            bb.y = ip[(size_t)(i + 1) * N] * okf;
            acc = __builtin_amdgcn_wmma_f32_16x16x4_f32(
                      false, a, false, bb, (short)0, acc, false, false);
        }
    }

    const int qq = q0 + col;
    float* ob = out + (size_t)b * outBatchStride;
#pragma unroll
    for (int r = 0; r < 8; ++r) {
        int v = r + 8 * hi;
        if (v < Cout && qq < N)
            ob[(size_t)v * N + qq] = acc[r] + bias[v];
    }
}

// ---------------------------------------------------------------------------
// Antialiased bilinear downsample of gt_mem (352x352 -> HxW), softmax over
// the HxW positions, +1.0 reweight. One block per batch image.
// ---------------------------------------------------------------------------
__global__ void mask_resize_softmax(const float* __restrict__ gt,
                                    float* __restrict__ maskw,
                                    int H, int W)
{
    __shared__ float vals[1936];   // max N = 44*44
    __shared__ float red[256];
    const int b   = blockIdx.x;
    const int tid = threadIdx.x;
    const int N   = H * W;
    const float s = 352.0f / (float)H;   // downscale factor (8/16/32)
    const float* g = gt + (size_t)b * 352 * 352;

    float lmax = -3.0e38f;
    for (int q = tid; q < N; q += 256) {
        int x = q % W, y = q / W;
        float cy = (y + 0.5f) * s - 0.5f;
        float cx = (x + 0.5f) * s - 0.5f;
        int y0 = (int)floorf(cy - s) + 1;
        int x0 = (int)floorf(cx - s) + 1;
        int taps = (int)(2.0f * s) + 1;
        float sum = 0.0f, wsum = 0.0f;
        for (int jy = 0; jy < taps; ++jy) {
            int sy = y0 + jy;
            float wy = 1.0f - fabsf((float)sy - cy) / s;
            if (wy <= 0.0f) continue;
            int syc = sy < 0 ? 0 : (sy > 351 ? 351 : sy);
            for (int jx = 0; jx < taps; ++jx) {
                int sx = x0 + jx;
                float wx = 1.0f - fabsf((float)sx - cx) / s;
                if (wx <= 0.0f) continue;
                int sxc = sx < 0 ? 0 : (sx > 351 ? 351 : sx);
                float wgt = wy * wx;
                sum  += wgt * g[syc * 352 + sxc];
                wsum += wgt;
            }
        }
        float v = sum / wsum;
        vals[q] = v;
        lmax = fmaxf(lmax, v);
    }
    red[tid] = lmax;
    __syncthreads();
    for (int off = 128; off > 0; off >>= 1) {
        if (tid < off) red[tid] = fmaxf(red[tid], red[tid + off]);
        __syncthreads();
    }
    float gmax = red[0];
    __syncthreads();

    float lsum = 0.0f;
    for (int q = tid; q < N; q += 256) {
        float e = __expf(vals[q] - gmax);
        vals[q] = e;
        lsum += e;
    }
    red[tid] = lsum;
    __syncthreads();
    for (int off = 128; off > 0; off >>= 1) {
        if (tid < off) red[tid] += red[tid + off];
        __syncthreads();
    }
    float gsum = red[0];
    __syncthreads();

    float inv = 1.0f / gsum;
    for (int q = tid; q < N; q += 256)
        maskw[(size_t)b * N + q] = vals[q] * inv + 1.0f;
}

// ---------------------------------------------------------------------------
// One flash step over a 16-row memory tile. TAIL=true adds range clamps and
// -inf masking for the final partial tile; the hot path is clamp-free.
// ---------------------------------------------------------------------------
template <bool TAIL>
__device__ __forceinline__ void stm_step(
    int m0, int N, int col, int hi,
    const float* __restrict__ kmb, const float* __restrict__ vmb,
    const float* __restrict__ mwb,
    v2f bq, v8f& acc, float& Mv, float& Lv)
{
    int mr = m0 + col;
    if (TAIL) { if (mr > N - 1) mr = N - 1; }
    v2f ak;                                 // A: km^T tile (16 x 4)
    ak.x = kmb[(size_t)(2 * hi + 0) * N + mr];
    ak.y = kmb[(size_t)(2 * hi + 1) * N + mr];
    v8f z = {};
    v8f s = __builtin_amdgcn_wmma_f32_16x16x4_f32(
                false, ak, false, bq, (short)0, z, false, false);

    float p[8];
    float cmax = -3.0e38f;
#pragma unroll
    for (int r = 0; r < 8; ++r) {
        float sv = s[r] * 0.5f;             // 1/sqrt(ck) = 0.5
        if (TAIL) { int m = m0 + r + 8 * hi; if (m >= N) sv = -3.0e38f; }
        p[r] = sv;
        cmax = fmaxf(cmax, sv);
    }
    cmax = fmaxf(cmax, xor16(cmax));        // full-column max
    float Mn   = fmaxf(Mv, cmax);
    float corr = __expf(Mv - Mn);
    Mv = Mn;

    float lsum = 0.0f;
#pragma unroll
    for (int r = 0; r < 8; ++r) {
        p[r] = __expf(p[r] - Mn);
        lsum += p[r];                       // denominator: unscaled p
    }
    Lv = Lv * corr + (lsum + xor16(lsum));
#pragma unroll
    for (int r = 0; r < 8; ++r) acc[r] *= corr;

    // fold the vm mask-reweight into P rows (after denominator accumulation)
#pragma unroll
    for (int r = 0; r < 8; ++r) {
        int m = m0 + r + 8 * hi;
        if (TAIL) m = m < N ? m : N - 1;
        p[r] *= mwb[m];                     // masked rows have p==0 anyway
    }

    float psw[8];
#pragma unroll
    for (int r = 0; r < 8; ++r) psw[r] = xor16(p[r]);

    // read-GEMM: acc[v,q] += vm0[v, m0+4c..m0+4c+3] * P[4c..4c+3, q]
#pragma unroll
    for (int c = 0; c < 4; ++c) {
        int mk0 = m0 + 4 * c + 2 * hi;
        int mk1 = mk0 + 1;
        if (TAIL) {
            mk0 = mk0 > N - 1 ? N - 1 : mk0;
            mk1 = mk1 > N - 1 ? N - 1 : mk1;
        }
        v2f av;
        av.x = vmb[(size_t)col * N + mk0];
        av.y = vmb[(size_t)col * N + mk1];

        int  t     = 4 * c + 2 * hi;        // P row this lane supplies
        int  ri    = t & 7;
        bool local = ((t >> 3) == hi);
        v2f bp;
        bp.x = local ? p[ri]     : psw[ri];
        bp.y = local ? p[ri + 1] : psw[ri + 1];

        acc = __builtin_amdgcn_wmma_f32_16x16x4_f32(
                  false, av, false, bp, (short)0, acc, false, false);
    }
}

// ---------------------------------------------------------------------------
// Fused STM attention, flash-softmax over the memory axis.
// One wave32 owns one 16-query tile of one batch.
// ---------------------------------------------------------------------------
__global__ void __launch_bounds__(256)
flash_stm_attn(const float* __restrict__ kq, const float* __restrict__ km,
               const float* __restrict__ vm, const float* __restrict__ mw,
               float* __restrict__ out, int N, int nQT)
{
    const int lane = threadIdx.x & 31;
    const int wave = threadIdx.x >> 5;
    const int qt   = blockIdx.x * 8 + wave;
    const int b    = blockIdx.y;
    if (qt >= nQT) return;                 // wave-uniform exit

    const int q0  = qt * 16;
    const int col = lane & 15;
    const int hi  = lane >> 4;

    const float* kqb = kq + (size_t)b * 4 * N;
    const float* kmb = km + (size_t)b * 4 * N;
    const float* vmb = vm + (size_t)b * 16 * N;
    const float* mwb = mw + (size_t)b * N;

    // loop-invariant B operand: kq tile (4 x 16)
    int qc = q0 + col; if (qc > N - 1) qc = N - 1;
    v2f bq;
    bq.x = kqb[(size_t)(2 * hi + 0) * N + qc];
    bq.y = kqb[(size_t)(2 * hi + 1) * N + qc];

    v8f acc = {};
    float Mv = -3.0e38f;
    float Lv = 0.0f;

    const int nFull = N >> 4;
    for (int mt = 0; mt < nFull; ++mt)
        stm_step<false>(mt * 16, N, col, hi, kmb, vmb, mwb, bq, acc, Mv, Lv);
    if (N & 15)
        stm_step<true>(nFull * 16, N, col, hi, kmb, vmb, mwb, bq, acc, Mv, Lv);

    // normalize and write read channels (0..15) of the output level
    float invL = 1.0f / Lv;
    float* ob = out + (size_t)b * 32 * N;
    int qq = q0 + col;
#pragma unroll
    for (int r = 0; r < 8; ++r) {
        int v = r + 8 * hi;
        if (qq < N) ob[(size_t)v * N + qq] = acc[r] * invL;
    }
}

// ---------------------------------------------------------------------------
extern "C" void kernel_launch(void* const* d_in, const int* in_sizes, int n_in,
                              void* d_out, int out_size, void* d_ws, size_t ws_size,
                              hipStream_t stream)
{
    (void)in_sizes; (void)n_in; (void)out_size; (void)ws_size;
    // setup_inputs() dict order:
    // 0 fmap_cur0, 1 fmap_mem0, 2 fmap_cur1, 3 fmap_mem1, 4 fmap_cur2,
    // 5 fmap_mem2, 6 gt_mem, then per level: wk,bk,wv,bv
    const float* fcur[3] = {(const float*)d_in[0], (const float*)d_in[2], (const float*)d_in[4]};
    const float* fmem[3] = {(const float*)d_in[1], (const float*)d_in[3], (const float*)d_in[5]};
    const float* gt      = (const float*)d_in[6];
    const float* wk[3]   = {(const float*)d_in[7],  (const float*)d_in[11], (const float*)d_in[15]};
    const float* bk[3]   = {(const float*)d_in[8],  (const float*)d_in[12], (const float*)d_in[16]};
    const float* wv[3]   = {(const float*)d_in[9],  (const float*)d_in[13], (const float*)d_in[17]};
    const float* bv[3]   = {(const float*)d_in[10], (const float*)d_in[14], (const float*)d_in[18]};

    const int sizes[3] = {44, 22, 11};
    float* ws = (float*)d_ws;
    size_t wsOff = 0;
    float* outLv = (float*)d_out;

    for (int lv = 0; lv < 3; ++lv) {
        const int H = sizes[lv], W = sizes[lv], N = H * W;
        float* kqb   = ws + wsOff; wsOff += (size_t)BATCH * 4  * N;
        float* kmb   = ws + wsOff; wsOff += (size_t)BATCH * 4  * N;
        float* vmb   = ws + wsOff; wsOff += (size_t)BATCH * 16 * N;
        float* maskw = ws + wsOff; wsOff += (size_t)BATCH * N;

        const int nQT = (N + 15) / 16;
        dim3 grid((nQT + 7) / 8, BATCH);

        // projections via WMMA im2col (vq written directly to out ch 16..31)
        conv3x3_wmma<<<grid, 256, 0, stream>>>(
            fcur[lv], wk[lv], bk[lv], kqb, 4, H, W, 4 * N, nQT);
        conv3x3_wmma<<<grid, 256, 0, stream>>>(
            fmem[lv], wk[lv], bk[lv], kmb, 4, H, W, 4 * N, nQT);
        conv3x3_wmma<<<grid, 256, 0, stream>>>(
            fmem[lv], wv[lv], bv[lv], vmb, 16, H, W, 16 * N, nQT);
        conv3x3_wmma<<<grid, 256, 0, stream>>>(
            fcur[lv], wv[lv], bv[lv], outLv + (size_t)16 * N, 16, H, W, 32 * N, nQT);

        mask_resize_softmax<<<BATCH, 256, 0, stream>>>(gt, maskw, H, W);

        flash_stm_attn<<<grid, 256, 0, stream>>>(
            kqb, kmb, vmb, maskw, outLv, N, nQT);

        outLv += (size_t)BATCH * 32 * N;
    }
}